// NormalizeFromCenter_19318762897443
// MI455X (gfx1250) — compile-verified
//
#include <hip/hip_runtime.h>
#include <cstdint>
#include <cstddef>

// ---------------------------------------------------------------------------
// out[n,d] = in[n,d] * mul(d), mul(d) in {1, 1/640, 1/360}, fixed pattern.
// Pure streaming: 1.14 GB traffic -> HBM-bound (~49us @ 23.3 TB/s on MI455X).
// CDNA5 path: async global->LDS staging (ASYNCcnt pipeline) + NT stores.
// ---------------------------------------------------------------------------

#define FEAT   4346
#define PAIRF  (2 * FEAT)      // 8692 floats per row-pair; 34768 B, 16B-aligned
#define PVEC   (PAIRF / 4)     // 2173 float4 per row-pair
#define PPB    16              // row-pairs per block (inner pipelined loop)
#define DEPTH  4               // async pipeline depth (per-wave)
#define BLOCK  256

typedef float v4f __attribute__((ext_vector_type(4)));
typedef int   v4i __attribute__((ext_vector_type(4)));

// Branchless multiplier: per 369-slot block (x2): half_w slots at
// {81,164,172+5j}, half_h at {82,165,173+5j}; everything from 738 up is 1.0.
__device__ __forceinline__ float mul_of(int d) {
  const float iw = 1.0f / 640.0f;  // 1 / (IM_W/2)
  const float ih = 1.0f / 360.0f;  // 1 / (IM_H/2)
  int  r   = (d >= 369) ? (d - 369) : d;   // v_cndmask
  int  t   = r - 172;
  unsigned q = (unsigned)t % 5u;           // valid only when t>=0 (masked below)
  bool obj = (t >= 0);
  bool w   = (r == 81) | (r == 164) | (obj & (q == 0u));
  bool h   = (r == 82) | (r == 165) | (obj & (q == 1u));
  bool in2 = (d < 738);
  float m = 1.0f;
  m = (in2 & w) ? iw : m;
  m = (in2 & h) ? ih : m;
  return m;
}

#if defined(__has_builtin)
#if __has_builtin(__builtin_amdgcn_global_load_async_to_lds_b128) && \
    __has_builtin(__builtin_amdgcn_s_wait_asynccnt)
#define USE_ASYNC 1
#endif
#endif
#ifndef USE_ASYNC
#define USE_ASYNC 0
#endif

#if USE_ASYNC
#pragma message("gfx1250: async global->LDS path ENABLED")
// Builtin params are pointers to int4: global side addrspace(1), LDS side
// addrspace(3) (verified by round-1 diagnostics + round-2 successful codegen).
typedef __attribute__((address_space(1))) v4i glb_v4i;
typedef __attribute__((address_space(3))) v4i lds_v4i;

__global__ __launch_bounds__(BLOCK) void
norm_async_kernel(const float* __restrict__ in, float* __restrict__ out,
                  int npairs) {
  __shared__ v4f stage[DEPTH][BLOCK];
  const int tid   = threadIdx.x;
  const int p     = blockIdx.x * BLOCK + tid;       // float4 index in row-pair
  const int pair0 = blockIdx.y * PPB;
  int jmax = npairs - pair0;
  if (jmax > PPB) jmax = PPB;
  if (p >= PVEC) return;                            // no barriers used -> safe

  // Column positions are invariant across the j loop: compute multiplier once.
  v4f m;
#pragma unroll
  for (int i = 0; i < 4; ++i) {
    int d = 4 * p + i;
    if (d >= FEAT) d -= FEAT;                       // element may sit in row 2
    m[i] = mul_of(d);
  }

  const size_t base = (size_t)pair0 * PAIRF + (size_t)(4 * p);

  // cpol=1 -> TH=NT (input is touched exactly once; don't pollute L2)
#define ISSUE(j, s)                                                         \
  __builtin_amdgcn_global_load_async_to_lds_b128(                           \
      (glb_v4i*)(in + base + (size_t)(j) * PAIRF),                          \
      (lds_v4i*)&stage[(s)][tid], 0, 1)

  const int pre = (jmax < DEPTH) ? jmax : DEPTH;
  for (int j = 0; j < pre; ++j) ISSUE(j, j);

  for (int j = 0; j < jmax; ++j) {
    if (j + DEPTH < jmax) {
      ISSUE(j + DEPTH, (j + DEPTH) & (DEPTH - 1));
      // issued = j+1+DEPTH; in-order completion => cnt<=DEPTH means chunk j done
      __builtin_amdgcn_s_wait_asynccnt(DEPTH);
    } else {
      __builtin_amdgcn_s_wait_asynccnt(0);          // drain at pipeline tail
    }
    asm volatile("" ::: "memory");                  // fence LDS read vs wait
    v4f v = stage[j & (DEPTH - 1)][tid];            // ds_load_b128
    v4f r = v * m;
    __builtin_nontemporal_store(r, (v4f*)(out + base + (size_t)j * PAIRF));
  }
#undef ISSUE
}

#else  // !USE_ASYNC ------------------------------------------------------
#pragma message("gfx1250: async builtins NOT found, direct NT path")

__global__ __launch_bounds__(BLOCK) void
norm_direct_kernel(const float* __restrict__ in, float* __restrict__ out,
                   int npairs) {
  const int tid   = threadIdx.x;
  const int p     = blockIdx.x * BLOCK + tid;
  const int pair0 = blockIdx.y * PPB;
  int jmax = npairs - pair0;
  if (jmax > PPB) jmax = PPB;
  if (p >= PVEC) return;

  v4f m;
#pragma unroll
  for (int i = 0; i < 4; ++i) {
    int d = 4 * p + i;
    if (d >= FEAT) d -= FEAT;
    m[i] = mul_of(d);
  }

  const size_t base = (size_t)pair0 * PAIRF + (size_t)(4 * p);
#pragma unroll 4
  for (int j = 0; j < jmax; ++j) {
    v4f v = __builtin_nontemporal_load((const v4f*)(in + base + (size_t)j * PAIRF));
    __builtin_nontemporal_store(v * m, (v4f*)(out + base + (size_t)j * PAIRF));
  }
}
#endif // USE_ASYNC

extern "C" void kernel_launch(void* const* d_in, const int* in_sizes, int n_in,
                              void* d_out, int out_size, void* d_ws, size_t ws_size,
                              hipStream_t stream) {
  (void)n_in; (void)out_size; (void)d_ws; (void)ws_size;
  const float* in  = (const float*)d_in[0];
  float*       out = (float*)d_out;

  const int total  = in_sizes[0];        // 32768 * 4346 = 142,409,728
  const int nrows  = total / FEAT;       // 32768
  const int npairs = nrows / 2;          // 16384 row-pairs (16B-aligned units)

  dim3 block(BLOCK);
  dim3 grid((PVEC + BLOCK - 1) / BLOCK,  // 9 blocks across a row-pair
            (npairs + PPB - 1) / PPB);   // 1024 pair-groups

#if USE_ASYNC
  hipLaunchKernelGGL(norm_async_kernel, grid, block, 0, stream, in, out, npairs);
#else
  hipLaunchKernelGGL(norm_direct_kernel, grid, block, 0, stream, in, out, npairs);
#endif
}